// MSSSIML1Loss_6158983102792
// MI455X (gfx1250) — compile-verified
//
#include <hip/hip_runtime.h>
#include <cmath>

// ---------------- problem constants ----------------
#define HH 384
#define WW 384
#define BB 128
#define RPS 48          // output rows per strip (on the 384 grid)
#define STRIPS 8        // 8 * 48 = 384
#define NWAVES 12       // 384 threads / wave32
#define HALO 4          // zeroed halo columns on each side of LDS rows
#define WP (WW + 2*HALO) // padded LDS row width = 392

typedef __attribute__((ext_vector_type(2))) float v2f;
typedef __attribute__((ext_vector_type(8))) float v8f;

// ---------------- wave32 sum reduction via the matrix engine ----------------
// A (16x4 f32): lanes 0-15 -> (M=lane,K=0|1) in vgpr0|1 ; lanes 16-31 -> (M,K=2|3).
// Value in component 0, zero in component 1; B = all ones.
// D[M][N] = v(lane M) + v(lane M+16); each lane sums its 8 D components
// (rows of its half), then one shfl_xor(16) completes the wave sum.
__device__ __forceinline__ float wave_reduce_sum(float v) {
#if __has_builtin(__builtin_amdgcn_wmma_f32_16x16x4_f32)
  v2f a;  a[0] = v;    a[1] = 0.0f;
  v2f bo; bo[0] = 1.0f; bo[1] = 1.0f;
  v8f c = {0.f, 0.f, 0.f, 0.f, 0.f, 0.f, 0.f, 0.f};
  v8f d = __builtin_amdgcn_wmma_f32_16x16x4_f32(
      /*neg_a=*/false, a, /*neg_b=*/false, bo,
      /*c_mod=*/(short)0, c, /*reuse_a=*/false, /*reuse_b=*/false);
  float p = d[0] + d[1] + d[2] + d[3] + d[4] + d[5] + d[6] + d[7];
  p += __shfl_xor(p, 16, 32);
  return p;
#else
  #pragma unroll
  for (int off = 16; off; off >>= 1) v += __shfl_xor(v, off, 32);
  return v;
#endif
}

// ---------------- one SSIM scale over one row-strip ----------------
// Box conv, zero padding p=K/2, weights 1/K^2. Output rows [i0,i1) on the
// (HH + 2p - K + 1)-row grid. vbuf: LDS, 2 phases x 5 maps x WP floats,
// halo columns pre-zeroed so the horizontal window needs NO bounds checks.
template <int K>
__device__ __forceinline__ void level_pass(const float* __restrict__ Xb,
                                           const float* __restrict__ Yb,
                                           int i0, int i1, float C1, float C2,
                                           float* __restrict__ vbuf,
                                           float& accS, float& accL1) {
  constexpr int   P    = K / 2;
  constexpr int   WOUT = WW + 2 * P - K + 1;   // 384 (k=1,7) or 385 (k=2,4)
  constexpr float INV  = 1.0f / (float)(K * K);
  const int t = threadIdx.x;                    // one thread per image column

  // zero the halo columns of both phase buffers (5 maps x 8 entries x 2 phases)
  if (t < 2 * 5 * 8) {
    const int ph  = t / 40;
    const int map = (t % 40) / 8;
    const int e   = t % 8;
    const int col = (e < HALO) ? e : (WP - 8 + e);   // 0..3 or 388..391
    vbuf[ph * (5 * WP) + map * WP + col] = 0.0f;
  }
  // visibility of the halo zeros is guaranteed by the per-row barrier below,
  // which always precedes the first halo read.

  float rx[K] = {};                             // ring of last K input rows
  float ry[K] = {};

  const int rb = i0 - P;                        // first input row needed
  const int re = (i1 - 1) + (K - 1) - P;        // last input row needed

  for (int r = rb; r <= re; ++r) {
    const int rr = r - rb;

    float x = 0.0f, y = 0.0f;                   // zero padding outside image
    if ((unsigned)r < (unsigned)HH) {
      const int idx = r * WW + t;
      x = Xb[idx];
      y = Yb[idx];
    }
    // gfx1250 global_prefetch_b8 of the next row
    if (r < re && (unsigned)(r + 1) < (unsigned)HH) {
      __builtin_prefetch(Xb + (r + 1) * WW + t, 0, 3);
      __builtin_prefetch(Yb + (r + 1) * WW + t, 0, 3);
    }

    // shift ring (fully unrolled -> stays in VGPRs)
#pragma unroll
    for (int q = 0; q < K - 1; ++q) { rx[q] = rx[q + 1]; ry[q] = ry[q + 1]; }
    rx[K - 1] = x;
    ry[K - 1] = y;

    if (rr >= K - 1) {                          // uniform across the block
      // vertical window sums for output row i = r + P - K + 1
      float sx = 0, sy = 0, sxx = 0, syy = 0, sxy = 0;
#pragma unroll
      for (int q = 0; q < K; ++q) {
        const float a = rx[q], b = ry[q];
        sx += a; sy += b; sxx += a * a; syy += b * b; sxy += a * b;
      }
      float* vb = vbuf + (rr & 1) * (5 * WP);
      vb[0 * WP + HALO + t] = sx;  vb[1 * WP + HALO + t] = sy;
      vb[2 * WP + HALO + t] = sxx; vb[3 * WP + HALO + t] = syy;
      vb[4 * WP + HALO + t] = sxy;
      __syncthreads();

      // horizontal window + SSIM; no bounds checks (halo columns are zero).
      // thread 0 also covers column 384 when WOUT == 385.
      for (int j = t; j < WOUT; j += WW) {
        const int c0 = j - P + HALO;
        float hx = 0, hy = 0, hxx = 0, hyy = 0, hxy = 0;
#pragma unroll
        for (int q = 0; q < K; ++q) {
          hx  += vb[0 * WP + c0 + q];
          hy  += vb[1 * WP + c0 + q];
          hxx += vb[2 * WP + c0 + q];
          hyy += vb[3 * WP + c0 + q];
          hxy += vb[4 * WP + c0 + q];
        }
        const float ux = hx * INV, uy = hy * INV;
        const float uxx = hxx * INV, uyy = hyy * INV, uxy = hxy * INV;
        const float vx = uxx - ux * ux, vy = uyy - uy * uy, vxy = uxy - ux * uy;
        const float A1 = 2.0f * ux * uy + C1, A2 = 2.0f * vxy + C2;
        const float B1 = ux * ux + uy * uy + C1, B2 = vx + vy + C2;
        accS += (A1 * A2) / (B1 * B2);
        if (K == 7) accL1 += fabsf(ux - uy);    // L1 term rides the k=7 pass
      }
    }
  }
}

// ---------------- main fused kernel ----------------
__global__ void __launch_bounds__(384)
msssim_main(const float* __restrict__ X, const float* __restrict__ Y,
            const float* __restrict__ dr, float* __restrict__ partials) {
  __shared__ float vbuf[2 * 5 * WP];            // ~15.7 KB double buffer
  __shared__ float wred[NWAVES * 8];

  const int blk = blockIdx.x;
  const int b = blk / STRIPS;
  const int s = blk % STRIPS;

  const float* Xb = X + (size_t)b * HH * WW;
  const float* Yb = Y + (size_t)b * HH * WW;

  const float d  = dr[b];
  const float C1 = (0.01f * d) * (0.01f * d);
  const float C2 = (0.03f * d) * (0.03f * d);

  const int i0  = s * RPS;
  const int i1a = i0 + RPS;                                  // 384-row grids
  const int i1b = (s == STRIPS - 1) ? (HH + 1) : (i0 + RPS); // 385-row grids

  float acc1 = 0, acc2 = 0, acc4 = 0, acc7 = 0, accL = 0;

  level_pass<1>(Xb, Yb, i0, i1a, C1, C2, vbuf, acc1, accL);
  __syncthreads();
  level_pass<2>(Xb, Yb, i0, i1b, C1, C2, vbuf, acc2, accL);
  __syncthreads();
  level_pass<4>(Xb, Yb, i0, i1b, C1, C2, vbuf, acc4, accL);
  __syncthreads();
  level_pass<7>(Xb, Yb, i0, i1a, C1, C2, vbuf, acc7, accL);

  // deterministic block reduction: wmma wave-sum -> LDS across 12 waves
  float vals[5] = {acc1, acc2, acc4, acc7, accL};
  const int lane = threadIdx.x & 31;
  const int wave = threadIdx.x >> 5;
#pragma unroll
  for (int m = 0; m < 5; ++m) {
    const float ws = wave_reduce_sum(vals[m]);
    if (lane == 0) wred[wave * 8 + m] = ws;
  }
  __syncthreads();
  if (threadIdx.x < 5) {
    float tot = 0.0f;
    for (int w = 0; w < NWAVES; ++w) tot += wred[w * 8 + threadIdx.x];
    partials[(size_t)blk * 5 + threadIdx.x] = tot;
  }
}

// ---------------- finalize: combine partials into the scalar loss ----------------
__global__ void msssim_finalize(const float* __restrict__ partials, int nblocks,
                                float* __restrict__ out) {
  __shared__ float red[256];
  float loc[5] = {0, 0, 0, 0, 0};
  for (int i = threadIdx.x; i < nblocks; i += 256) {
#pragma unroll
    for (int m = 0; m < 5; ++m) loc[m] += partials[(size_t)i * 5 + m];
  }
  float tot[5];
#pragma unroll
  for (int m = 0; m < 5; ++m) {
    red[threadIdx.x] = loc[m];
    __syncthreads();
    for (int off = 128; off; off >>= 1) {
      if ((int)threadIdx.x < off) red[threadIdx.x] += red[threadIdx.x + off];
      __syncthreads();
    }
    tot[m] = red[0];
    __syncthreads();
  }
  if (threadIdx.x == 0) {
    const double N384 = (double)BB * 384.0 * 384.0;
    const double N385 = (double)BB * 385.0 * 385.0;
    const double m1 = tot[0] / N384;
    const double m2 = tot[1] / N385;
    const double m4 = tot[2] / N385;
    const double m7 = tot[3] / N384;
    const double msssim = m1 * m2 * m4 * m7;
    const double l1 = tot[4] / N384;
    out[0] = (float)(0.84 * (1.0 - msssim) + 0.16 * l1);
  }
}

// ---------------- launch ----------------
extern "C" void kernel_launch(void* const* d_in, const int* in_sizes, int n_in,
                              void* d_out, int out_size, void* d_ws, size_t ws_size,
                              hipStream_t stream) {
  (void)in_sizes; (void)n_in; (void)out_size; (void)ws_size;
  const float* X  = (const float*)d_in[0];
  const float* Y  = (const float*)d_in[1];
  const float* dr = (const float*)d_in[2];
  float* partials = (float*)d_ws;       // 1024 blocks * 5 floats = 20 KB
  float* out      = (float*)d_out;

  const int nblocks = BB * STRIPS;      // 1024
  msssim_main<<<dim3(nblocks), dim3(384), 0, stream>>>(X, Y, dr, partials);
  msssim_finalize<<<dim3(1), dim3(256), 0, stream>>>(partials, nblocks, out);
}